// SLSTM_21131239096714
// MI455X (gfx1250) — compile-verified
//
#include <hip/hip_runtime.h>
#include <hip/hip_bf16.h>

// ---- CDNA5 WMMA fragment types (wave32) -----------------------------------
typedef __attribute__((ext_vector_type(2))) float v2f;  // A/B frag for 16x16x4 f32
typedef __attribute__((ext_vector_type(8))) float v8f;  // 16x16 f32 C/D frag

__device__ __forceinline__ float sigmoidf_(float x) {
  return 1.0f / (1.0f + __expf(-x));
}

// ---------------------------------------------------------------------------
// Kernel 1: gates GEMM (i,g,o only; f-gate is dead since c0==0) + activations
//   h_relu[b, j] = relu( sigmoid(o) * tanh( sigmoid(i) * tanh(g) ) )
//   gates = x[4096,1024] @ W_ih^T + (b_ih + b_hh)
// One wave computes a 32(M)x16(N) tile of h: 2 M-subtiles x 3 gates = 6
// accumulators. Per k-step: 2 A-frag loads + 3 B-frag loads feed 6 WMMAs
// (5 loads / 6 WMMAs vs 4/3 unblocked -> 9.6 FLOP per L2 byte).
// ---------------------------------------------------------------------------
__global__ __launch_bounds__(256) void slstm_gates_kernel(
    const float* __restrict__ x,      // [4096, 1024]
    const float* __restrict__ W_ih,   // [2048, 1024] row-major
    const float* __restrict__ b_ih,   // [2048]
    const float* __restrict__ b_hh,   // [2048]
    float* __restrict__ h_relu)       // [4096, 512] workspace
{
  const int IN = 1024, Hh = 512;

  const int w    = threadIdx.x >> 5;       // wave in block (8 waves)
  const int lane = threadIdx.x & 31;
  const int gw   = blockIdx.x * 8 + w;     // global wave id
  const int mtile = gw >> 5;               // 128 M-supertiles (4096/32)
  const int ntile = gw & 31;               // 32  N-tiles (512/16)
  const int row0 = mtile * 32;
  const int col0 = ntile * 16;

  const int lx    = lane & 15;             // N (for B/C/D) or M (for A)
  const int khalf = lane >> 4;             // 0: K=0,1   1: K=2,3

  // A frags: lane holds x[row + lx, k + 2*khalf + {0,1}]  (contiguous pair)
  const float* aptr0 = x + (size_t)(row0 +  0 + lx) * IN + 2 * khalf;
  const float* aptr1 = x + (size_t)(row0 + 16 + lx) * IN + 2 * khalf;
  // B frags: lane holds W_ih[gate*512 + col0 + lx, k + 2*khalf + {0,1}]
  const float* bptr_i = W_ih + (size_t)(col0 +    0 + lx) * IN + 2 * khalf;
  const float* bptr_g = W_ih + (size_t)(col0 + 1024 + lx) * IN + 2 * khalf;
  const float* bptr_o = W_ih + (size_t)(col0 + 1536 + lx) * IN + 2 * khalf;

  v8f acc_i0 = {}, acc_g0 = {}, acc_o0 = {};
  v8f acc_i1 = {}, acc_g1 = {}, acc_o1 = {};

  for (int k = 0; k < IN; k += 4) {
    v2f a0 = *(const v2f*)(aptr0  + k);
    v2f a1 = *(const v2f*)(aptr1  + k);
    v2f bi = *(const v2f*)(bptr_i + k);
    v2f bg = *(const v2f*)(bptr_g + k);
    v2f bo = *(const v2f*)(bptr_o + k);
    acc_i0 = __builtin_amdgcn_wmma_f32_16x16x4_f32(false, a0, false, bi,
                                                   (short)0, acc_i0, false, false);
    acc_i1 = __builtin_amdgcn_wmma_f32_16x16x4_f32(false, a1, false, bi,
                                                   (short)0, acc_i1, false, false);
    acc_g0 = __builtin_amdgcn_wmma_f32_16x16x4_f32(false, a0, false, bg,
                                                   (short)0, acc_g0, false, false);
    acc_g1 = __builtin_amdgcn_wmma_f32_16x16x4_f32(false, a1, false, bg,
                                                   (short)0, acc_g1, false, false);
    acc_o0 = __builtin_amdgcn_wmma_f32_16x16x4_f32(false, a0, false, bo,
                                                   (short)0, acc_o0, false, false);
    acc_o1 = __builtin_amdgcn_wmma_f32_16x16x4_f32(false, a1, false, bo,
                                                   (short)0, acc_o1, false, false);
  }

  // Bias depends only on the output column (lane-uniform across the 8 VGPRs).
  const int jn = col0 + lx;
  const float bias_i = b_ih[jn +    0] + b_hh[jn +    0];
  const float bias_g = b_ih[jn + 1024] + b_hh[jn + 1024];
  const float bias_o = b_ih[jn + 1536] + b_hh[jn + 1536];

  // C/D layout: VGPR v, lane L  ->  M = v + 8*(L>=16), N = L&15
#pragma unroll
  for (int v = 0; v < 8; ++v) {
    {
      float gi = acc_i0[v] + bias_i;
      float gg = acc_g0[v] + bias_g;
      float go = acc_o0[v] + bias_o;
      float cn = sigmoidf_(gi) * tanhf(gg);   // c0 == 0 -> f-gate term vanishes
      float h  = sigmoidf_(go) * tanhf(cn);
      int m = row0 + v + 8 * khalf;
      h_relu[(size_t)m * Hh + jn] = fmaxf(h, 0.0f);
    }
    {
      float gi = acc_i1[v] + bias_i;
      float gg = acc_g1[v] + bias_g;
      float go = acc_o1[v] + bias_o;
      float cn = sigmoidf_(gi) * tanhf(gg);
      float h  = sigmoidf_(go) * tanhf(cn);
      int m = row0 + 16 + v + 8 * khalf;
      h_relu[(size_t)m * Hh + jn] = fmaxf(h, 0.0f);
    }
  }
}

// ---------------------------------------------------------------------------
// Kernel 2: fused head.
//   o1 = relu(h) @ W1^T + b1   -> [4096, 128]
//   o2 = relu(o1) @ W2^T + b2  -> [4096, 1]
//   out[s, b, 0] = o2[b]  for s in [0, num_steps)
// One wave computes a 16-row block: 8 accumulators span all 128 features; the
// W2 dot is folded into the epilogue via an intra-half shuffle reduction.
// ---------------------------------------------------------------------------
__global__ __launch_bounds__(256) void slstm_head_kernel(
    const float* __restrict__ h_relu, // [4096, 512]
    const float* __restrict__ W1,     // [128, 512]
    const float* __restrict__ b1,     // [128]
    const float* __restrict__ W2,     // [1, 128]
    const float* __restrict__ b2,     // [1]
    float* __restrict__ out,          // [num_steps, 4096]
    int num_steps)
{
  const int Hh = 512, Bn = 4096;

  const int w    = threadIdx.x >> 5;
  const int lane = threadIdx.x & 31;
  const int mtile = blockIdx.x * 8 + w;    // 256 M-tiles
  const int row0 = mtile * 16;

  const int lx    = lane & 15;
  const int khalf = lane >> 4;

  const float* aptr  = h_relu + (size_t)(row0 + lx) * Hh + 2 * khalf;
  const float* bbase = W1     + (size_t)lx * Hh + 2 * khalf;

  v8f acc[8];
#pragma unroll
  for (int t = 0; t < 8; ++t) acc[t] = (v8f){};

  for (int k = 0; k < Hh; k += 4) {
    v2f a = *(const v2f*)(aptr + k);
#pragma unroll
    for (int t = 0; t < 8; ++t) {
      v2f b = *(const v2f*)(bbase + (size_t)t * 16 * Hh + k);
      acc[t] = __builtin_amdgcn_wmma_f32_16x16x4_f32(false, a, false, b,
                                                     (short)0, acc[t], false, false);
    }
  }

  float w2v[8], b1v[8];
#pragma unroll
  for (int t = 0; t < 8; ++t) {
    w2v[t] = W2[t * 16 + lx];
    b1v[t] = b1[t * 16 + lx];
  }
  const float bias2 = b2[0];

#pragma unroll
  for (int v = 0; v < 8; ++v) {
    // partial over this lane's 8 feature columns for row M = v + 8*khalf
    float p = 0.0f;
#pragma unroll
    for (int t = 0; t < 8; ++t)
      p += fmaxf(acc[t][v] + b1v[t], 0.0f) * w2v[t];

    // reduce across the 16 lanes of this half (xor masks stay within halves)
    p += __shfl_xor(p, 1, 32);
    p += __shfl_xor(p, 2, 32);
    p += __shfl_xor(p, 4, 32);
    p += __shfl_xor(p, 8, 32);

    if (lx == 0) {
      const int m = row0 + v + 8 * khalf;
      const float val = p + bias2;
      for (int s = 0; s < num_steps; ++s)
        out[(size_t)s * Bn + m] = val;   // broadcast across time steps
    }
  }
}

// ---------------------------------------------------------------------------
extern "C" void kernel_launch(void* const* d_in, const int* in_sizes, int n_in,
                              void* d_out, int out_size, void* d_ws, size_t ws_size,
                              hipStream_t stream) {
  (void)in_sizes; (void)n_in; (void)ws_size;

  const float* x    = (const float*)d_in[0];
  const float* W_ih = (const float*)d_in[1];
  const float* b_ih = (const float*)d_in[2];
  // d_in[3] = W_hh: provably unused (h0 == 0, only its bias survives)
  const float* b_hh = (const float*)d_in[4];
  const float* W1   = (const float*)d_in[5];
  const float* b1   = (const float*)d_in[6];
  const float* W2   = (const float*)d_in[7];
  const float* b2   = (const float*)d_in[8];
  // d_in[9] = num_steps (device scalar); derive on host from out_size instead.
  const int Bn = 4096;
  const int num_steps = out_size / Bn;

  float* h_relu = (float*)d_ws;          // 4096*512*4 = 8 MB scratch
  float* out    = (float*)d_out;

  // 128 M-supertiles * 32 N-tiles = 4096 waves, 8 waves/block
  slstm_gates_kernel<<<512, 256, 0, stream>>>(x, W_ih, b_ih, b_hh, h_relu);
  // 256 M-tiles, 8 waves/block
  slstm_head_kernel<<<32, 256, 0, stream>>>(h_relu, W1, b1, W2, b2, out, num_steps);
}